// CrfLayer_14508399526225
// MI455X (gfx1250) — compile-verified
//
#include <hip/hip_runtime.h>
#include <cstddef>

// Problem constants (match reference)
#define NL 32          // labels
#define NT 1024        // time steps
#define NB 512         // batch
#define SOS 0
#define END_LBL 1
#define NCH 8          // time chunks per batch (scan split)
#define CS (NT / NCH)  // 128 steps per chunk
#define RSC 4          // rescale period (steps); CS % RSC == 0
#define PITCH 34       // LDS column pitch (words): even (8B align) + bank spread
#define WSLOT 1088     // per-wave LDS slab in floats (>= 31*34+32)

typedef __attribute__((ext_vector_type(2))) float v2f;
typedef __attribute__((ext_vector_type(8))) float v8f;

__device__ __forceinline__ float wave_max(float x) {
#pragma unroll
  for (int off = 16; off > 0; off >>= 1) x = fmaxf(x, __shfl_xor(x, off, 32));
  return x;
}
__device__ __forceinline__ float wave_sum(float x) {
#pragma unroll
  for (int off = 16; off > 0; off >>= 1) x += __shfl_xor(x, off, 32);
  return x;
}

// ---------------------------------------------------------------------------
// K1: per (batch, chunk) wave computes the 32x32 chunk transfer matrix
//     G_c = prod_{t in chunk} (diag(exp(feat_t)) * M),  M = exp(transitions).
//     Every RSC steps the matrix max is reduced and a power-of-two scale
//     2^-k is folded (exactly) into the next step's diag; k accumulates.
//     Core op: D = A(16x4) x B(4x16) + C via v_wmma_f32_16x16x4_f32.
// ---------------------------------------------------------------------------
__global__ __launch_bounds__(256)
void crf_fwd_chunk(const float* __restrict__ emis,
                   const float* __restrict__ trans,
                   float* __restrict__ Gout,
                   float* __restrict__ lsOut)
{
  __shared__ __align__(16) float lds[8 * WSLOT];
  const int lane = threadIdx.x & 31;
  const int wib  = threadIdx.x >> 5;          // wave in block (8 waves)
  const int gw   = blockIdx.x * 8 + wib;      // global wave = (b, c)
  const int b    = gw >> 3;
  const int c    = gw & (NCH - 1);
  float* g = &lds[wib * WSLOT];               // this wave's G slab (col-major, pitch 34)

  const int h  = lane >> 4;                   // lane half: K offset 2h
  const int ln = lane & 15;

  // A-fragments of M = exp(transitions): tile jt in {0,1}, k-slice q in 0..7.
  // A layout (f32 16x4): lane row = ln (+16*jt tile), K = 4q + 2h + {0,1}.
  v2f A[2][8];
#pragma unroll
  for (int jt = 0; jt < 2; ++jt) {
    const int row = ln + 16 * jt;
#pragma unroll
    for (int q = 0; q < 8; ++q) {
      const int k = 4 * q + 2 * h;
      v2f a;
      a.x = __expf(trans[row * NL + k]);
      a.y = __expf(trans[row * NL + k + 1]);
      A[jt][q] = a;
    }
  }

  // G := Identity in LDS (column `lane`), col-major: g[col*PITCH + rowI]
#pragma unroll 4
  for (int m = 0; m < NL; ++m)
    g[lane * PITCH + m] = (m == lane) ? 1.0f : 0.0f;

  int   ktot = 0;      // accumulated power-of-two rescale exponent
  float sc   = 1.0f;   // 2^-k scale to fold into this step's diag (then reset)
  const float* ep = emis + ((size_t)b * NT + (size_t)c * CS) * NL + lane;
  float feat = ep[0];

  v8f Cc[2][2];

#pragma unroll 1
  for (int t = 0; t < CS; ++t) {
    const float featn = (t + 1 < CS) ? ep[(size_t)(t + 1) * NL] : 0.0f; // prefetch
    const float e = __expf(feat) * sc;       // D_t, scaled once after a rescale
    sc = 1.0f;
    const float d0 = __shfl(e, ln, 32);      // exp(feat[row]) for rows 0..15
    const float d1 = __shfl(e, ln + 16, 32); // rows 16..31

    // B-fragments of previous G from LDS: B[K = 4q+2h+{0,1}][N = ln+16kt]
    v2f Bf[8][2];
#pragma unroll
    for (int q = 0; q < 8; ++q)
#pragma unroll
      for (int kt = 0; kt < 2; ++kt)
        Bf[q][kt] = *(const v2f*)&g[(ln + 16 * kt) * PITCH + 4 * q + 2 * h];

#pragma unroll
    for (int jt = 0; jt < 2; ++jt)
#pragma unroll
      for (int kt = 0; kt < 2; ++kt) {
        v8f z;
#pragma unroll
        for (int i = 0; i < 8; ++i) z[i] = 0.0f;
        Cc[jt][kt] = z;
      }

    // G_new = (D*M) * G_old : 32 wmma per step, tiles interleaved to hide latency
#pragma unroll
    for (int q = 0; q < 8; ++q) {
      const v2f As0 = A[0][q] * d0;   // row-scale M by D (rows = A's M dim)
      const v2f As1 = A[1][q] * d1;
      Cc[0][0] = __builtin_amdgcn_wmma_f32_16x16x4_f32(false, As0, false, Bf[q][0], (short)0, Cc[0][0], false, false);
      Cc[0][1] = __builtin_amdgcn_wmma_f32_16x16x4_f32(false, As0, false, Bf[q][1], (short)0, Cc[0][1], false, false);
      Cc[1][0] = __builtin_amdgcn_wmma_f32_16x16x4_f32(false, As1, false, Bf[q][0], (short)0, Cc[1][0], false, false);
      Cc[1][1] = __builtin_amdgcn_wmma_f32_16x16x4_f32(false, As1, false, Bf[q][1], (short)0, Cc[1][1], false, false);
    }

    // periodic power-of-two rescale (folded into next step's diag)
    if (((t & (RSC - 1)) == (RSC - 1)) && (t != CS - 1)) {
      float mx = Cc[0][0][0];
#pragma unroll
      for (int jt = 0; jt < 2; ++jt)
#pragma unroll
        for (int kt = 0; kt < 2; ++kt)
#pragma unroll
          for (int i = 0; i < 8; ++i) mx = fmaxf(mx, Cc[jt][kt][i]);
      mx = wave_max(mx);
      // mx is a positive normal float: exponent k s.t. mx * 2^-k in [0.5, 1)
      const int k = (__float_as_int(mx) >> 23) - 126;
      ktot += k;
      sc = __int_as_float((127 - k) << 23);  // exact 2^-k
    }

    // write G_new to LDS (col-major): row j = r+8h+16jt contiguous in r -> b64
#pragma unroll
    for (int jt = 0; jt < 2; ++jt)
#pragma unroll
      for (int kt = 0; kt < 2; ++kt)
#pragma unroll
        for (int r = 0; r < 8; r += 2) {
          v2f st;
          st.x = Cc[jt][kt][r];
          st.y = Cc[jt][kt][r + 1];
          *(v2f*)&g[(ln + 16 * kt) * PITCH + 16 * jt + 8 * h + r] = st;
        }
    feat = featn;
  }

  // store chunk matrix (row-major [j][i]) + accumulated logscale (= ktot*ln2)
  float* go = Gout + (size_t)gw * (NL * NL);
#pragma unroll
  for (int jt = 0; jt < 2; ++jt)
#pragma unroll
    for (int kt = 0; kt < 2; ++kt)
#pragma unroll
      for (int r = 0; r < 8; ++r) {
        const int j   = r + 8 * h + 16 * jt;
        const int col = ln + 16 * kt;
        go[j * NL + col] = Cc[jt][kt][r];
      }
  if (lane == 0) lsOut[gw] = (float)ktot * 0.6931471805599453f;
}

// ---------------------------------------------------------------------------
// K2: gold path score per batch (deterministic block tree reduction)
// ---------------------------------------------------------------------------
__global__ __launch_bounds__(256)
void crf_gold(const float* __restrict__ emis, const int* __restrict__ lab,
              const float* __restrict__ trans, float* __restrict__ gold)
{
  const int b = blockIdx.x, tid = threadIdx.x;
  float part = 0.0f;
  for (int t = tid; t < NT; t += 256) {
    const int tag  = lab[(size_t)b * NT + t];
    const int prev = (t == 0) ? SOS : lab[(size_t)b * NT + t - 1];
    part += trans[tag * NL + prev] + emis[((size_t)b * NT + t) * NL + tag];
    if (t == NT - 1) part += trans[END_LBL * NL + tag];
  }
  __shared__ float red[256];
  red[tid] = part; __syncthreads();
#pragma unroll
  for (int s = 128; s > 0; s >>= 1) {
    if (tid < s) red[tid] += red[tid + s];
    __syncthreads();
  }
  if (tid == 0) gold[b] = red[0];
}

// ---------------------------------------------------------------------------
// K3: per-batch sequential combine over the 8 chunk matrices (one wave/batch),
//     then forward_score - gold_score
// ---------------------------------------------------------------------------
__global__ __launch_bounds__(32)
void crf_combine(const float* __restrict__ G, const float* __restrict__ ls,
                 const float* __restrict__ trans, const float* __restrict__ gold,
                 float* __restrict__ diff)
{
  const int b = blockIdx.x, j = threadIdx.x;  // j = label lane
  float v   = (j == SOS) ? 1.0f : 0.0f;       // exp(alpha0)
  float acc = 0.0f;
  for (int c = 0; c < NCH; ++c) {
    const float* g = G + ((size_t)(b * NCH + c)) * (NL * NL);
    float w = 0.0f;
#pragma unroll
    for (int i = 0; i < NL; ++i)
      w += g[j * NL + i] * __shfl(v, i, 32);  // v = G_c * v
    const float m = wave_max(w);
    v = w / m;
    acc += __logf(m) + ls[b * NCH + c];
  }
  const float s = wave_sum(v * __expf(trans[END_LBL * NL + j]));
  if (j == 0) diff[b] = __logf(s) + acc - gold[b];  // forward - gold
}

// ---------------------------------------------------------------------------
// K4: deterministic final reduction over batches -> scalar loss
// ---------------------------------------------------------------------------
__global__ __launch_bounds__(512)
void crf_reduce(const float* __restrict__ diff, float* __restrict__ out)
{
  __shared__ float red[NB];
  const int tid = threadIdx.x;
  red[tid] = diff[tid];
  __syncthreads();
#pragma unroll
  for (int s = NB / 2; s > 0; s >>= 1) {
    if (tid < s) red[tid] += red[tid + s];
    __syncthreads();
  }
  if (tid == 0) out[0] = red[0];
}

extern "C" void kernel_launch(void* const* d_in, const int* in_sizes, int n_in,
                              void* d_out, int out_size, void* d_ws, size_t ws_size,
                              hipStream_t stream) {
  (void)in_sizes; (void)n_in; (void)out_size; (void)ws_size;
  const float* emis  = (const float*)d_in[0];
  const int*   lab   = (const int*)d_in[1];
  const float* trans = (const float*)d_in[2];

  float* w    = (float*)d_ws;
  float* G    = w;                                   // 512*8*1024 f32 = 16 MiB
  float* ls   = G + (size_t)NB * NCH * NL * NL;      // 4096 f32
  float* gold = ls + NB * NCH;                       // 512 f32
  float* diff = gold + NB;                           // 512 f32

  crf_fwd_chunk<<<NB, 256, 0, stream>>>(emis, trans, G, ls);  // 4096 waves
  crf_gold    <<<NB, 256, 0, stream>>>(emis, lab, trans, gold);
  crf_combine <<<NB, 32, 0, stream>>>(G, ls, trans, gold, diff);
  crf_reduce  <<<1, NB, 0, stream>>>(diff, (float*)d_out);
}